// HashEncoderNativeLegacy_3530463117845
// MI455X (gfx1250) — compile-verified
//
#include <hip/hip_runtime.h>

// ---------------------------------------------------------------------------
// 4D multiresolution hash-grid encode (instant-NGP style), MI455X / gfx1250.
//
// Gather-bound kernel: 67M random float2 gathers into a 64MB table that is
// resident in the 192MB L2. One wave32 lane handles one (point, level) pair:
//   - 16 independent global_load_b64 gathers (hash table, RT-cached)
//   - quadrilinear lerp tree in VALU
//   - coalesced non-temporal b64 store (don't pollute L2 with the 32MB output)
// Point coords are staged block-wide through LDS with the CDNA5 async
// global->LDS path (ASYNCcnt) when the builtin is available.
// ---------------------------------------------------------------------------

#ifndef __has_builtin
#define __has_builtin(x) 0
#endif

#if defined(__AMDGCN__) && __has_builtin(__builtin_amdgcn_global_load_async_to_lds_b128)
#define USE_ASYNC_LDS 1
#else
#define USE_ASYNC_LDS 0
#endif

#define NLEV   16u
#define TSIZE  (1u << 19)          // LOG2_HASHMAP_SIZE = 19
#define TMASK  (TSIZE - 1u)

typedef float v2f __attribute__((ext_vector_type(2)));
typedef float v4f __attribute__((ext_vector_type(4)));
typedef int   v4i __attribute__((vector_size(16)));   // matches builtin param

#if USE_ASYNC_LDS
typedef __attribute__((address_space(1))) v4i* g_v4i_p;   // global (device) AS
typedef __attribute__((address_space(3))) v4i* l_v4i_p;   // LDS AS
#endif

// floor(MIN_RES * growth^l), growth = 2^(4/15)  (matches the numpy float64 table)
__constant__ float kScale[16] = {
    16.f,  19.f,  23.f,  27.f,  33.f,  40.f,  48.f,  58.f,
    70.f,  84.f, 101.f, 122.f, 147.f, 176.f, 212.f, 256.f};

__global__ __launch_bounds__(256) void hashgrid_encode_kernel(
    const v4f* __restrict__ xyzt,   // [N,4]
    const v2f* __restrict__ tbl,    // [TSIZE*NLEV, 2]
    v2f* __restrict__ out)          // [N, NLEV] float2  == [N, 32] float
{
    const unsigned t     = blockIdx.x * 256u + threadIdx.x;
    const unsigned level = threadIdx.x & 15u;          // 16 lanes share a point

    // ---- stage this block's 16 points' coords (16 x 16B) into LDS ----------
#if USE_ASYNC_LDS
    __shared__ v4f pts[16];
    if (threadIdx.x < 16u) {
        const v4f* gsrc = xyzt + (blockIdx.x * 16u + threadIdx.x);
        // generic -> integer -> AS-qualified pointer (clang forbids direct
        // pointer casts that change address spaces; the LDS offset is the low
        // 32 bits of the generic address per the aperture mapping).
        g_v4i_p gp = (g_v4i_p)(unsigned long long)gsrc;
        l_v4i_p lp = (l_v4i_p)(unsigned)(unsigned long long)(&pts[threadIdx.x]);
        __builtin_amdgcn_global_load_async_to_lds_b128(gp, lp, 0, 0);
    }
#if __has_builtin(__builtin_amdgcn_s_wait_asynccnt)
    __builtin_amdgcn_s_wait_asynccnt(0);
#else
    asm volatile("s_wait_asynccnt 0" ::: "memory");
#endif
    __syncthreads();
    const v4f cd = pts[threadIdx.x >> 4];
#else
    const v4f cd = xyzt[t >> 4];
#endif

    // ---- per-level scaling, corner coords, lerp weights --------------------
    const float sc = kScale[level];
    const float sx = cd.x * sc, sy = cd.y * sc, sz = cd.z * sc, sw = cd.w * sc;
    const float fx = __builtin_floorf(sx), fy = __builtin_floorf(sy);
    const float fz = __builtin_floorf(sz), fw = __builtin_floorf(sw);
    const float wx = sx - fx, wy = sy - fy, wz = sz - fz, ww = sw - fw;

    // hashes: uint32 wrapping prime products (prime0 == 1), matching reference
    const unsigned P1 = 2654435761u, P2 = 805459861u, P3 = 3674653429u;
    const unsigned hf0 = (unsigned)fx;
    const unsigned hf1 = (unsigned)fy * P1;
    const unsigned hf2 = (unsigned)fz * P2;
    const unsigned hf3 = (unsigned)fw * P3;
    const unsigned hc0 = (unsigned)__builtin_ceilf(sx);
    const unsigned hc1 = (unsigned)__builtin_ceilf(sy) * P1;
    const unsigned hc2 = (unsigned)__builtin_ceilf(sz) * P2;
    const unsigned hc3 = (unsigned)__builtin_ceilf(sw) * P3;

    const unsigned base = level << 19;   // HASH_OFFSET = level * TSIZE

    // ---- 16 independent corner gathers (L2-resident table) -----------------
    v2f g[16];
#pragma unroll
    for (unsigned c = 0; c < 16u; ++c) {
        const unsigned h = ((c & 1u) ? hc0 : hf0) ^ ((c & 2u) ? hc1 : hf1)
                         ^ ((c & 4u) ? hc2 : hf2) ^ ((c & 8u) ? hc3 : hf3);
        g[c] = tbl[base + (h & TMASK)];
    }

    // ---- quadrilinear reduction, reference order (dim0/bit0 first) ---------
    v2f r8[8];
#pragma unroll
    for (int i = 0; i < 8; ++i) r8[i] = g[2*i]  * (1.f - wx) + g[2*i+1]  * wx;
    v2f r4[4];
#pragma unroll
    for (int i = 0; i < 4; ++i) r4[i] = r8[2*i] * (1.f - wy) + r8[2*i+1] * wy;
    v2f r2[2];
#pragma unroll
    for (int i = 0; i < 2; ++i) r2[i] = r4[2*i] * (1.f - wz) + r4[2*i+1] * wz;
    const v2f r = r2[0] * (1.f - ww) + r2[1] * ww;

    // ---- coalesced streaming store (NT: keep L2 for the hash table) --------
    __builtin_nontemporal_store(r, out + t);
}

extern "C" void kernel_launch(void* const* d_in, const int* in_sizes, int n_in,
                              void* d_out, int out_size, void* d_ws, size_t ws_size,
                              hipStream_t stream) {
    (void)n_in; (void)out_size; (void)d_ws; (void)ws_size;
    const v4f* xyzt = (const v4f*)d_in[0];          // N x 4 float
    const v2f* tbl  = (const v2f*)d_in[1];          // (TSIZE*NLEV) x 2 float
    v2f*       out  = (v2f*)d_out;                  // N x 16 float2

    const unsigned npts     = (unsigned)(in_sizes[0] / 4);
    const unsigned nthreads = npts * NLEV;          // 4,194,304 for N=262144
    const unsigned blocks   = nthreads / 256u;      // exact: N divisible by 16

    hashgrid_encode_kernel<<<dim3(blocks), dim3(256), 0, stream>>>(xyzt, tbl, out);
}